// WavetableSynth_60112362274919
// MI455X (gfx1250) — compile-verified
//
#include <hip/hip_runtime.h>
#include <math.h>
#include <stdint.h>

// ---------------- problem constants (from reference setup_inputs) -----------
#define B_ROWS   64
#define N_SAMP   65536
#define FRAMES   256
#define SR_INV   (1.0f / 48000.0f)
#define EPS_STAB 1e-3f
#define NPOS     64
#define NWT      2048
#define TPB      256
#define STAGE    8192          // samples per stage
#define PER_T    32            // STAGE / TPB
#define NSTAGES  (N_SAMP / STAGE)
#define INV_TWO_PI 0.15915493667125702f   // 1/(2*pi) in fp32

typedef float v2f  __attribute__((ext_vector_type(2)));
typedef float v8f  __attribute__((ext_vector_type(8)));
typedef unsigned int u32x4 __attribute__((ext_vector_type(4)));
typedef int   i32x4 __attribute__((ext_vector_type(4)));
typedef int   i32x8 __attribute__((ext_vector_type(8)));

#if __has_builtin(__builtin_amdgcn_tensor_load_to_lds)
#define HAS_TDM 1
#else
#define HAS_TDM 0
#endif

#if __has_builtin(__builtin_amdgcn_wmma_f32_16x16x4_f32)
#define HAS_WMMA_F32 1
#else
#define HAS_WMMA_F32 0
#endif

// ---------------- CDNA5 TDM: 1-D tile of f32 from global -> LDS -------------
__device__ __forceinline__ void tdm_issue_f0(const float* gsrc, float* ldsdst,
                                             int nelem) {
#if HAS_TDM
  unsigned long long ga = (unsigned long long)(uintptr_t)gsrc;
  u32x4 g0;
  g0.x = 1u;                                    // count = 1 valid descriptor
  g0.y = (unsigned)(uintptr_t)ldsdst;           // lds byte address (low 32 bits of flat LDS ptr)
  g0.z = (unsigned)ga;                          // global_addr[31:0]
  g0.w = ((unsigned)(ga >> 32) & 0x01FFFFFFu)   // global_addr[56:32]
         | (2u << 30);                          // type = 2 ("image")
  i32x8 g1 = {0, 0, 0, 0, 0, 0, 0, 0};
  g1[0] = (int)(2u << 16);                      // data_size = 2 -> 4 bytes
  g1[1] = (int)(((unsigned)nelem & 0xFFFFu) << 16);     // tensor_dim0 lo16
  g1[2] = (int)((((unsigned)nelem >> 16) & 0xFFFFu)     // tensor_dim0 hi16
                | (1u << 16));                          // tensor_dim1 = 1
  g1[3] = (int)(((unsigned)nelem & 0xFFFFu) << 16);     // tile_dim0 = nelem
  g1[5] = nelem;                                // tensor_dim0_stride lo32
  i32x4 z4 = {0, 0, 0, 0};
#if __has_include(<hip/amd_detail/amd_gfx1250_TDM.h>)
  i32x8 z8 = {0, 0, 0, 0, 0, 0, 0, 0};          // clang-23 / therock: 6-arg form
  __builtin_amdgcn_tensor_load_to_lds(g0, g1, z4, z4, z8, 0);
#else
  __builtin_amdgcn_tensor_load_to_lds(g0, g1, z4, z4, 0);  // ROCm 7.2: 5-arg form
#endif
#else
  (void)gsrc; (void)ldsdst; (void)nelem;
#endif
}

__device__ __forceinline__ void wait_tdm() {
#if __has_builtin(__builtin_amdgcn_s_wait_tensorcnt)
  __builtin_amdgcn_s_wait_tensorcnt(0);
#else
  asm volatile("s_wait_tensorcnt 0x0" ::: "memory");
#endif
}

// linear interp of a frame-rate control padded to [0..FRAMES] (257 entries)
__device__ __forceinline__ float finterp(const float* fr, int i) {
  int   fi = i / 257;                         // exact: N-1 = 255*257
  float t  = (float)(i - fi * 257) * (1.0f / 257.0f);
  float v0 = fr[fi];
  return v0 + (fr[fi + 1] - v0) * t;
}

__global__ void __launch_bounds__(TPB)
synth_kernel(const float* __restrict__ f0, const float* __restrict__ phase,
             const float* __restrict__ add_lfo, const float* __restrict__ sub,
             const float* __restrict__ env, const float* __restrict__ wt,
             float* __restrict__ out) {
  extern __shared__ float smem[];
  const int b   = blockIdx.x;
  const int tid = threadIdx.x;

  // ---- LDS carving -------------------------------------------------------
  float* f0buf  = smem;                  // 2 * STAGE   (double buffer; retired half reused as wet staging)
  float* wtf    = f0buf + 2 * STAGE;     // 257
  float* envf   = wtf  + 257;
  float* a1f    = envf + 257;
  float* a2f    = a1f  + 257;
  float* b0f    = a2f  + 257;
  float* b1f    = b0f  + 257;
  float* b2f    = b1f  + 257;
  float* totals = b2f  + 257;            // 256 phase partials
  float* rp     = totals + 256;          // 256 scan scratch
  float* aff    = rp + 256;              // 256 * 6 affine IIR scan
  float* carry  = aff + 256 * 6;         // [0]=phase carry [1]=y1 [2]=y2 [3]=stage total

  // ---- once per block: frame-rate controls -> LDS ------------------------
  {
    const int f = tid;
    float al = add_lfo[b * FRAMES + f];
    float ev = env[b * FRAMES + f];
    const float* s5 = sub + (size_t)(b * FRAMES + f) * 5;
    float l0 = s5[0], l1 = s5[1];
    const float stab = 1.0f - EPS_STAB;
    float a1 = 2.0f * tanhf(l0) * stab;
    float aa = fabsf(a1);
    float a2 = ((2.0f - aa) * tanhf(l1) * stab + aa) * 0.5f;
    wtf[f] = al * 2.0f - 1.0f;
    envf[f] = ev;  a1f[f] = a1;  a2f[f] = a2;
    b0f[f] = s5[2];  b1f[f] = s5[3];  b2f[f] = s5[4];
    if (f == FRAMES - 1) {               // pad so fi+1 never reads OOB
      wtf[FRAMES] = al * 2.0f - 1.0f;  envf[FRAMES] = ev;
      a1f[FRAMES] = a1;  a2f[FRAMES] = a2;
      b0f[FRAMES] = s5[2];  b1f[FRAMES] = s5[3];  b2f[FRAMES] = s5[4];
    }
    if (tid == 0) {
      carry[0] = phase[b] * INV_TWO_PI;  // phase/(2*pi)
      carry[1] = 0.0f;  carry[2] = 0.0f;  carry[3] = 0.0f;
    }
  }

  // warm the wavetable (512 KB, lives in L2 for the whole kernel)
  __builtin_prefetch(wt + tid * 512, 0, 1);

  const float* f0row  = f0 + (size_t)b * N_SAMP;
  float*       outrow = out + (size_t)b * N_SAMP;

  // ---- stage 0 f0 load ---------------------------------------------------
#if HAS_TDM
  if (tid == 0) { tdm_issue_f0(f0row, f0buf, STAGE); wait_tdm(); }
#else
#pragma unroll
  for (int k = 0; k < PER_T; ++k) f0buf[tid + TPB * k] = f0row[tid + TPB * k];
#endif
  __syncthreads();

  for (int s = 0; s < NSTAGES; ++s) {
    float* cur = f0buf + (s & 1) * STAGE;
    float* nxt = f0buf + ((s + 1) & 1) * STAGE;
    const int base = s * STAGE;
    const int c0   = tid * PER_T;

#if HAS_TDM
    if (s + 1 < NSTAGES && tid == 0)
      tdm_issue_f0(f0row + base + STAGE, nxt, STAGE);   // overlap DMA with compute
#endif

    // ---- phase: thread-local partial sums -------------------------------
    float lsum = 0.0f;
#pragma unroll
    for (int j = 0; j < PER_T; ++j) lsum += cur[c0 + j] * SR_INV;
    totals[tid] = lsum;
    __syncthreads();

    // ---- block scan of 256 partials -------------------------------------
#if HAS_WMMA_F32
    if (tid < 32) {                       // wave 0, EXEC all ones
      const int L = tid;
      const int m = L & 15;               // A row
      const int ko = (L >> 4) * 2;        // A K pair within slice
      const int n = L & 15;               // B/D column
      const int kl = (L >> 4);            // B K within slice
      v8f acc = {0, 0, 0, 0, 0, 0, 0, 0};
#pragma unroll
      for (int ks = 0; ks < 4; ++ks) {    // 16x16 = 4 chained K=4 WMMAs
        v2f a, bm;
        a.x = totals[16 * m + 4 * ks + ko];
        a.y = totals[16 * m + 4 * ks + ko + 1];
        bm.x = (4 * ks + kl     <= n) ? 1.0f : 0.0f;   // upper-tri ones
        bm.y = (4 * ks + kl + 2 <= n) ? 1.0f : 0.0f;
        acc = __builtin_amdgcn_wmma_f32_16x16x4_f32(
            false, a, false, bm, (short)0, acc, false, false);
      }
#pragma unroll
      for (int v = 0; v < 8; ++v) {       // D layout: M = v + 8*(L>>4), N = L&15
        rp[(v + 8 * (L >> 4)) * 16 + n] = acc[v];   // row-wise inclusive prefix
      }
    }
#else
    rp[tid] = totals[tid];
    __syncthreads();
    for (int off = 1; off < TPB; off <<= 1) {
      float v = rp[tid];
      float a = (tid >= off) ? rp[tid - off] : 0.0f;
      __syncthreads();
      rp[tid] = v + a;
      __syncthreads();
    }
#endif
    __syncthreads();

    float ex;                              // exclusive prefix of totals
#if HAS_WMMA_F32
    {
      int r = tid >> 4, c = tid & 15;
      float within = (c > 0) ? rp[r * 16 + (c - 1)] : 0.0f;
      float rowoff = 0.0f;
      for (int rr = 0; rr < r; ++rr) rowoff += rp[rr * 16 + 15];
      ex = within + rowoff;
    }
#else
    ex = (tid > 0) ? rp[tid - 1] : 0.0f;
#endif
    if (tid == TPB - 1) carry[3] = ex + lsum;   // stage phase total
    const float ph0 = carry[0] + ex;

    // ---- dry = wavetable_osc * env (32 samples in registers) ------------
    float dryv[PER_T];
    float run = ph0;
#pragma unroll
    for (int j = 0; j < PER_T; ++j) {
      const int i = base + c0 + j;
      run += cur[c0 + j] * SR_INV;
      float u    = run - floorf(run);          // mod 1
      float samp = u * (float)NWT;
      float sfl  = floorf(samp);
      float sf   = samp - sfl;
      int s0i = ((int)sfl) & (NWT - 1);
      int s1i = (s0i + 1) & (NWT - 1);
      float wtv = finterp(wtf, i);
      float ev  = finterp(envf, i);
      float pos = fminf(fmaxf((wtv + 1.0f) * 0.5f, 0.0f), 1.0f) * (float)(NPOS - 1);
      float pfl = floorf(pos);
      float pf  = pos - pfl;
      int p0 = (int)pfl;
      int p1 = (p0 + 1 < NPOS - 1) ? (p0 + 1) : (NPOS - 1);
      const float* r0 = wt + p0 * NWT;
      const float* r1 = wt + p1 * NWT;
      float w00 = r0[s0i], w01 = r0[s1i], w10 = r1[s0i], w11 = r1[s1i];
      float top = w00 + (w01 - w00) * sf;
      float bot = w10 + (w11 - w10) * sf;
      dryv[j] = (top + (bot - top) * pf) * ev;
    }

    // ---- IIR: compose chunk affine transform (state = (y[n-1], y[n-2])) --
    float A00 = 1.0f, A01 = 0.0f, A10 = 0.0f, A11 = 1.0f, C0 = 0.0f, C1 = 0.0f;
#pragma unroll
    for (int j = 0; j < PER_T; ++j) {
      const int i = base + c0 + j;
      float a1 = finterp(a1f, i);
      float a2 = finterp(a2f, i);
      float n00 = -a1 * A00 - a2 * A10;
      float n01 = -a1 * A01 - a2 * A11;
      A10 = A00;  A11 = A01;  A00 = n00;  A01 = n01;
      float nc0 = -a1 * C0 - a2 * C1 + dryv[j];
      C1 = C0;  C0 = nc0;
    }
    {
      float* W = aff + tid * 6;
      W[0] = A00; W[1] = A01; W[2] = A10; W[3] = A11; W[4] = C0; W[5] = C1;
    }
    __syncthreads();
    if (tid == 0) carry[0] = carry[0] + carry[3];   // phase carry for next stage

    // ---- Hillis-Steele composition scan over 256 chunk transforms --------
    for (int off = 1; off < TPB; off <<= 1) {
      float q0 = aff[tid * 6 + 0], q1 = aff[tid * 6 + 1], q2 = aff[tid * 6 + 2],
            q3 = aff[tid * 6 + 3], q4 = aff[tid * 6 + 4], q5 = aff[tid * 6 + 5];
      float p0 = 0, p1 = 0, p2 = 0, p3 = 0, p4 = 0, p5 = 0;
      if (tid >= off) {
        const float* P = aff + (tid - off) * 6;
        p0 = P[0]; p1 = P[1]; p2 = P[2]; p3 = P[3]; p4 = P[4]; p5 = P[5];
      }
      __syncthreads();
      if (tid >= off) {                 // new = q ∘ p  (p first, then q)
        float* W = aff + tid * 6;
        W[0] = q0 * p0 + q1 * p2;
        W[1] = q0 * p1 + q1 * p3;
        W[2] = q2 * p0 + q3 * p2;
        W[3] = q2 * p1 + q3 * p3;
        W[4] = q0 * p4 + q1 * p5 + q4;
        W[5] = q2 * p4 + q3 * p5 + q5;
      }
      __syncthreads();
    }

    // ---- incoming state per chunk, replay exactly, fuse FIR -------------
    float cy1 = carry[1], cy2 = carry[2];
    float y1, y2;
    if (tid == 0) {
      y1 = cy1;  y2 = cy2;
    } else {
      const float* P = aff + (tid - 1) * 6;
      y1 = P[0] * cy1 + P[1] * cy2 + P[4];
      y2 = P[2] * cy1 + P[3] * cy2 + P[5];
    }
#pragma unroll
    for (int j = 0; j < PER_T; ++j) {
      const int i = base + c0 + j;
      float a1 = finterp(a1f, i);
      float a2 = finterp(a2f, i);
      float bb0 = finterp(b0f, i);
      float bb1 = finterp(b1f, i);
      float bb2 = finterp(b2f, i);
      float y = dryv[j] - a1 * y1 - a2 * y2;
      cur[c0 + j] = bb0 * y + bb1 * y1 + bb2 * y2;   // wet -> staging (f0 slice retired)
      y2 = y1;  y1 = y;
    }
    __syncthreads();
    if (tid == TPB - 1) { carry[1] = y1;  carry[2] = y2; }

    // ---- coalesced wet store --------------------------------------------
#pragma unroll
    for (int k = 0; k < PER_T; ++k)
      outrow[base + tid + TPB * k] = cur[tid + TPB * k];

#if HAS_TDM
    if (s + 1 < NSTAGES && tid == 0) wait_tdm();
#else
    if (s + 1 < NSTAGES) {
#pragma unroll
      for (int k = 0; k < PER_T; ++k)
        nxt[tid + TPB * k] = f0row[base + STAGE + tid + TPB * k];
    }
#endif
    __syncthreads();
  }
}

extern "C" void kernel_launch(void* const* d_in, const int* in_sizes, int n_in,
                              void* d_out, int out_size, void* d_ws, size_t ws_size,
                              hipStream_t stream) {
  (void)in_sizes; (void)n_in; (void)out_size; (void)d_ws; (void)ws_size;
  const float* f0      = (const float*)d_in[0];  // (64, 65536)
  const float* phase   = (const float*)d_in[1];  // (64, 1)
  const float* add_lfo = (const float*)d_in[2];  // (64, 256, 1)
  const float* sub     = (const float*)d_in[3];  // (64, 256, 5)
  const float* env     = (const float*)d_in[4];  // (64, 256)
  const float* wt      = (const float*)d_in[5];  // (64, 2048)
  float* out = (float*)d_out;                    // wet (64, 65536)

  const size_t shmem =
      (size_t)(2 * STAGE + 7 * 257 + 256 + 256 + 256 * 6 + 8) * sizeof(float);
  synth_kernel<<<B_ROWS, TPB, shmem, stream>>>(f0, phase, add_lfo, sub, env, wt,
                                               out);
}